// MOTIF_CORD_22634477650492
// MI455X (gfx1250) — compile-verified
//
#include <hip/hip_runtime.h>
#include <math.h>

// ---------------------------------------------------------------------------
// Problem constants (from the reference)
// ---------------------------------------------------------------------------
#define PHN   4
#define CHN   8
#define ZF    8          // full z
#define ZE    6          // effective z slices (z = 1..6)
#define HH    320
#define WW    320
#define K2    (2 * WW)   // real-K of the complexified GEMM (640)
#define NK    4096
#define HWSZ  (HH * WW)
#define NSLICE (PHN * CHN * ZE)          // 192 k-space slices
#define GAMMA 0.1f
#define TAU   0.2f
#define EPSW  0.01f
#define NITER 3

typedef float v2f __attribute__((ext_vector_type(2)));
typedef float v8f __attribute__((ext_vector_type(8)));

__device__ __forceinline__ float clampf(float v, float lo, float hi) {
    return fminf(fmaxf(v, lo), hi);
}

// ---------------------------------------------------------------------------
// Generic zero-fill (graph-capture-safe replacement for memset)
// ---------------------------------------------------------------------------
__global__ void k_fill(float* __restrict__ p, long long n) {
    long long i = (long long)blockIdx.x * blockDim.x + threadIdx.x;
    long long stride = (long long)gridDim.x * blockDim.x;
    for (; i < n; i += stride) p[i] = 0.0f;
}

// ---------------------------------------------------------------------------
// DFT twiddle matrices: D[n][k] = exp(-2*pi*i*n*k/320), Dc = conj(D)
// ---------------------------------------------------------------------------
__global__ void k_twiddle(float* __restrict__ D, float* __restrict__ Dc) {
    int i = blockIdx.x * blockDim.x + threadIdx.x;
    if (i >= HH * WW) return;
    int n = i / WW, k = i % WW;
    int m = (n * k) % WW;                       // exact phase wrap
    float ang = -(6.283185307179586f / (float)WW) * (float)m;
    float s, c;
    __sincosf(ang, &s, &c);
    D[2 * i] = c;  D[2 * i + 1] = s;
    Dc[2 * i] = c; Dc[2 * i + 1] = -s;
}

// ---------------------------------------------------------------------------
// x init: nan_to_num(image_init) -> xcur (PH, ZF, H, W) complex
// ---------------------------------------------------------------------------
__global__ void k_init_x(const float* __restrict__ img, float* __restrict__ xcur,
                         long long n2) {
    long long i = (long long)blockIdx.x * blockDim.x + threadIdx.x;
    if (i >= n2) return;
    float v = img[i];
    if (isnan(v)) v = 0.0f;
    else if (isinf(v)) v = (v > 0.0f) ? 3.4028235e38f : -3.4028235e38f;
    xcur[i] = v;
}

// ---------------------------------------------------------------------------
// Forward: motion warp + coil multiply
// xcur(PH,ZF,H,W)c, mvf(PH,2,ZF,H,W), csm(CH,ZF,H,W)c -> coil(slice,H,W)c
// slice = (ph*CH + ch)*ZE + ze, z = ze+1
// ---------------------------------------------------------------------------
__global__ void k_warp_coil(const float* __restrict__ xcur,
                            const float* __restrict__ mvf,
                            const float* __restrict__ csm,
                            float* __restrict__ coil) {
    int idx = blockIdx.x * blockDim.x + threadIdx.x;
    if (idx >= PHN * ZE * HWSZ) return;
    int ph = idx / (ZE * HWSZ);
    int r  = idx % (ZE * HWSZ);
    int ze = r / HWSZ;
    int p  = r % HWSZ;
    int z  = ze + 1;
    int y  = p / WW, x = p % WW;

    float f0 = mvf[((size_t)(ph * 2 + 0) * ZF + z) * HWSZ + p];
    float f1 = mvf[((size_t)(ph * 2 + 1) * ZF + z) * HWSZ + p];
    float gy = clampf((float)y + f0, 0.0f, (float)(HH - 1));
    float gx = clampf((float)x + f1, 0.0f, (float)(WW - 1));
    float fy0 = floorf(gy), fx0 = floorf(gx);
    float wy = gy - fy0, wx = gx - fx0;
    int y0 = (int)fy0, x0 = (int)fx0;
    int y1 = min(y0 + 1, HH - 1), x1 = min(x0 + 1, WW - 1);
    float w00 = (1.f - wy) * (1.f - wx), w01 = (1.f - wy) * wx;
    float w10 = wy * (1.f - wx),         w11 = wy * wx;

    const float* base = xcur + ((size_t)(ph * ZF + z) * HWSZ) * 2;
    float2 c00 = *(const float2*)(base + (size_t)(y0 * WW + x0) * 2);
    float2 c01 = *(const float2*)(base + (size_t)(y0 * WW + x1) * 2);
    float2 c10 = *(const float2*)(base + (size_t)(y1 * WW + x0) * 2);
    float2 c11 = *(const float2*)(base + (size_t)(y1 * WW + x1) * 2);
    float vr = c00.x * w00 + c01.x * w01 + c10.x * w10 + c11.x * w11;
    float vi = c00.y * w00 + c01.y * w01 + c10.y * w10 + c11.y * w11;

    #pragma unroll
    for (int ch = 0; ch < CHN; ++ch) {
        float2 cs = *(const float2*)(csm + ((size_t)(ch * ZF + z) * HWSZ + p) * 2);
        float orr = cs.x * vr - cs.y * vi;
        float oii = cs.x * vi + cs.y * vr;
        size_t s = ((size_t)(ph * CHN + ch) * ZE + ze);
        float2* o = (float2*)(coil + (s * HWSZ + p) * 2);
        *o = make_float2(orr, oii);
    }
}

// ---------------------------------------------------------------------------
// Batched DFT-GEMM via V_WMMA_F32_16X16X4_F32, complexified-real formulation:
//   The interleaved complex slice A (320x320c) IS a real matrix 320 x 640.
//   The DFT matrix Dw expands on the fly into a 640x16 real LDS block panel:
//       blk[2k  ][2c] =  re   blk[2k  ][2c+1] = im
//       blk[2k+1][2c] = -im   blk[2k+1][2c+1] = re
//   so  C = A_real x blk  directly yields interleaved complex A*Dw.
//   Inner step: 1 global b64 (A frag) + 1 ds 2addr_b32 (B frag) + 1 WMMA,
//   no shuffles, no negation VALU (sign baked into the panel).
// Block: 320 threads = 10 wave32 waves; each wave owns a 16-row x 8-ccol tile.
// grid = (2 row-stripes of 160, 40 complex-col groups of 8, batch=192).
// transStore: 1 -> store C transposed (complex), 0 -> normal orientation.
// ---------------------------------------------------------------------------
__global__ __launch_bounds__(320) void k_cgemm320(
    const float* __restrict__ A, long long strideA,
    const float* __restrict__ Dw,
    float* __restrict__ C, long long strideC,
    int transStore) {
    __shared__ float ldsB[K2 * 16];              // 40960 B real block panel

    const int slice = blockIdx.z;
    const float* Ab = A + (size_t)slice * strideA;
    float*       Cb = C + (size_t)slice * strideC;
    const int cc0 = blockIdx.y * 8;              // first complex output column
    const int tid = threadIdx.x;

    // Build the complex->real block panel in LDS (8 complex cols, all 320 k).
    for (int i = tid; i < WW * 8; i += 320) {
        int k = i >> 3, c = i & 7;
        float2 d = *(const float2*)(Dw + ((size_t)k * WW + cc0 + c) * 2);
        float* p0 = ldsB + (size_t)(2 * k) * 16 + 2 * c;
        p0[0]  = d.x;  p0[1]  = d.y;             // row 2k
        p0[16] = -d.y; p0[17] = d.x;             // row 2k+1
    }
    __syncthreads();

    const int wave = tid >> 5;
    const int lane = tid & 31;
    const int rowBase = blockIdx.x * 160 + wave * 16;
    const int n    = lane & 15;                  // real column within tile
    const int kh   = (lane >> 4) << 1;           // 0 (lanes 0-15) / 2 (16-31)
    const int half = lane >> 4;

    const float* ap = Ab + (size_t)(rowBase + n) * K2 + kh;  // A row, interleaved
    const float* lp = ldsB + (size_t)kh * 16 + n;
    v8f acc = {};

    for (int kk = 0; kk < K2; kk += 32) {
        __builtin_prefetch(ap + 64, 0, 3);       // next 256B of this A row
        #pragma unroll
        for (int u = 0; u < 8; ++u) {
            v2f a = *(const v2f*)(ap + u * 4);               // b64, contiguous
            v2f b = { lp[u * 64], lp[u * 64 + 16] };         // ds_load_2addr
            acc = __builtin_amdgcn_wmma_f32_16x16x4_f32(
                false, a, false, b, (short)0, acc, false, false);
        }
        ap += 32;
        lp += 32 * 16;
    }

    // C tile: VGPR v holds row rowBase + v + 8*half; real col n ->
    // complex col cc0 + (n>>1), parity n&1 selects re/im.
    const int cc  = cc0 + (n >> 1);
    const int par = n & 1;
    if (transStore) {
        float* cbase = Cb + (size_t)cc * WW * 2 + par;       // C^T layout
        #pragma unroll
        for (int v = 0; v < 8; ++v) {
            int row = rowBase + v + 8 * half;
            cbase[(size_t)row * 2] = acc[v];
        }
    } else {
        float* cbase = Cb + (size_t)cc * 2 + par;            // normal layout
        #pragma unroll
        for (int v = 0; v < 8; ++v) {
            int row = rowBase + v + 8 * half;
            cbase[(size_t)row * WW * 2] = acc[v];
        }
    }
}

// ---------------------------------------------------------------------------
// K-space bilinear sampler with folded fftshift:  Fs[y][x] = F[(y+160)%320][..]
// ---------------------------------------------------------------------------
__device__ __forceinline__ void traj_coords(const float* traj, int ph, int k,
                                            int& y0, int& y1, int& x0, int& x1,
                                            float& w00, float& w01,
                                            float& w10, float& w11) {
    const float inv2pi = 0.15915494309189535f;
    float ky = traj[((size_t)(ph * 2 + 0)) * NK + k];
    float kx = traj[((size_t)(ph * 2 + 1)) * NK + k];
    float gy = clampf((ky * inv2pi + 0.5f) * (float)HH, 0.0f, (float)(HH - 1));
    float gx = clampf((kx * inv2pi + 0.5f) * (float)WW, 0.0f, (float)(WW - 1));
    float fy0 = floorf(gy), fx0 = floorf(gx);
    float wy = gy - fy0, wx = gx - fx0;
    y0 = (int)fy0; x0 = (int)fx0;
    y1 = min(y0 + 1, HH - 1); x1 = min(x0 + 1, WW - 1);
    w00 = (1.f - wy) * (1.f - wx); w01 = (1.f - wy) * wx;
    w10 = wy * (1.f - wx);         w11 = wy * wx;
}

__device__ __forceinline__ int shift_idx(int y, int x) {
    return ((y + HH / 2) % HH) * WW + ((x + WW / 2) % WW);
}

__global__ void k_sample(const float* __restrict__ F,
                         const float* __restrict__ traj,
                         float* __restrict__ est) {
    int idx = blockIdx.x * blockDim.x + threadIdx.x;
    if (idx >= NSLICE * NK) return;
    int s = idx / NK, k = idx % NK;
    int ph = s / (CHN * ZE);
    int y0, y1, x0, x1; float w00, w01, w10, w11;
    traj_coords(traj, ph, k, y0, y1, x0, x1, w00, w01, w10, w11);
    const float* base = F + (size_t)s * HWSZ * 2;
    float2 c00 = *(const float2*)(base + (size_t)shift_idx(y0, x0) * 2);
    float2 c01 = *(const float2*)(base + (size_t)shift_idx(y0, x1) * 2);
    float2 c10 = *(const float2*)(base + (size_t)shift_idx(y1, x0) * 2);
    float2 c11 = *(const float2*)(base + (size_t)shift_idx(y1, x1) * 2);
    float er = c00.x * w00 + c01.x * w01 + c10.x * w10 + c11.x * w11;
    float ei = c00.y * w00 + c01.y * w01 + c10.y * w10 + c11.y * w11;
    *(float2*)(est + (size_t)idx * 2) = make_float2(er, ei);
}

// ---------------------------------------------------------------------------
// max |est|^2 reduction (block LDS tree + bit-pattern atomicMax)
// ---------------------------------------------------------------------------
__global__ void k_maxabs(const float* __restrict__ est, long long n,
                         float* __restrict__ maxsq) {
    __shared__ float red[256];
    long long i = (long long)blockIdx.x * blockDim.x + threadIdx.x;
    long long stride = (long long)gridDim.x * blockDim.x;
    float m = 0.0f;
    for (; i < n; i += stride) {
        float re = est[2 * i], im = est[2 * i + 1];
        m = fmaxf(m, re * re + im * im);
    }
    red[threadIdx.x] = m;
    __syncthreads();
    for (int s = 128; s > 0; s >>= 1) {
        if (threadIdx.x < s)
            red[threadIdx.x] = fmaxf(red[threadIdx.x], red[threadIdx.x + s]);
        __syncthreads();
    }
    if (threadIdx.x == 0)
        atomicMax((unsigned int*)maxsq, __float_as_uint(red[0]));
}

// ---------------------------------------------------------------------------
// Weighted residual: r = w^2 (est - y)/N,  w = 1/(|est|/max|est| + EPS)
// ---------------------------------------------------------------------------
__global__ void k_residual(const float* __restrict__ est,
                           const float* __restrict__ kdata,
                           const float* __restrict__ maxsq,
                           float* __restrict__ res) {
    int idx = blockIdx.x * blockDim.x + threadIdx.x;
    if (idx >= NSLICE * NK) return;
    int s = idx / NK, k = idx % NK;
    int ph = s / (CHN * ZE);
    int r2 = s % (CHN * ZE);
    int ch = r2 / ZE, ze = r2 % ZE;
    float2 e = *(const float2*)(est + (size_t)idx * 2);
    size_t yidx = (((size_t)(ph * CHN + ch)) * ZF + (ze + 1)) * NK + k;
    float2 yv = *(const float2*)(kdata + yidx * 2);
    float mag = sqrtf(e.x * e.x + e.y * e.y);
    float mm  = sqrtf(fmaxf(maxsq[0], 1e-30f));
    float w   = 1.0f / (mag / mm + EPSW);
    const float invN = 1.0f / (float)(NSLICE * NK);
    float sc = w * w * invN;
    *(float2*)(res + (size_t)idx * 2) =
        make_float2(sc * (e.x - yv.x), sc * (e.y - yv.y));
}

// ---------------------------------------------------------------------------
// Adjoint of the sampler: scatter residual into the (zeroed) TRANSPOSED
// k-space grid G^T (so the adjoint GEMMs keep the A-per-slice form).
// ---------------------------------------------------------------------------
__global__ void k_scatter(const float* __restrict__ res,
                          const float* __restrict__ traj,
                          float* __restrict__ GT) {
    int idx = blockIdx.x * blockDim.x + threadIdx.x;
    if (idx >= NSLICE * NK) return;
    int s = idx / NK, k = idx % NK;
    int ph = s / (CHN * ZE);
    int y0, y1, x0, x1; float w00, w01, w10, w11;
    traj_coords(traj, ph, k, y0, y1, x0, x1, w00, w01, w10, w11);
    float2 r = *(const float2*)(res + (size_t)idx * 2);
    float* base = GT + (size_t)s * HWSZ * 2;
    // transposed scatter: swap (y,x) roles in the shifted index
    int i00 = shift_idx(x0, y0), i01 = shift_idx(x1, y0);
    int i10 = shift_idx(x0, y1), i11 = shift_idx(x1, y1);
    atomicAdd(base + (size_t)i00 * 2,     w00 * r.x);
    atomicAdd(base + (size_t)i00 * 2 + 1, w00 * r.y);
    atomicAdd(base + (size_t)i01 * 2,     w01 * r.x);
    atomicAdd(base + (size_t)i01 * 2 + 1, w01 * r.y);
    atomicAdd(base + (size_t)i10 * 2,     w10 * r.x);
    atomicAdd(base + (size_t)i10 * 2 + 1, w10 * r.y);
    atomicAdd(base + (size_t)i11 * 2,     w11 * r.x);
    atomicAdd(base + (size_t)i11 * 2 + 1, w11 * r.y);
}

// ---------------------------------------------------------------------------
// Coil combine (conj csm, sum over CH) + adjoint of the motion warp (scatter)
// ---------------------------------------------------------------------------
__global__ void k_combine_warpadj(const float* __restrict__ Gimg,
                                  const float* __restrict__ csm,
                                  const float* __restrict__ mvf,
                                  float* __restrict__ gdc) {
    int idx = blockIdx.x * blockDim.x + threadIdx.x;
    if (idx >= PHN * ZE * HWSZ) return;
    int ph = idx / (ZE * HWSZ);
    int r  = idx % (ZE * HWSZ);
    int ze = r / HWSZ;
    int p  = r % HWSZ;
    int z  = ze + 1;
    int y  = p / WW, x = p % WW;

    float gr = 0.0f, gi = 0.0f;
    #pragma unroll
    for (int ch = 0; ch < CHN; ++ch) {
        size_t s = ((size_t)(ph * CHN + ch) * ZE + ze);
        float2 g  = *(const float2*)(Gimg + (s * HWSZ + p) * 2);
        float2 cs = *(const float2*)(csm + ((size_t)(ch * ZF + z) * HWSZ + p) * 2);
        gr += cs.x * g.x + cs.y * g.y;           // conj(csm) * g
        gi += cs.x * g.y - cs.y * g.x;
    }

    float f0 = mvf[((size_t)(ph * 2 + 0) * ZF + z) * HWSZ + p];
    float f1 = mvf[((size_t)(ph * 2 + 1) * ZF + z) * HWSZ + p];
    float gy = clampf((float)y + f0, 0.0f, (float)(HH - 1));
    float gx = clampf((float)x + f1, 0.0f, (float)(WW - 1));
    float fy0 = floorf(gy), fx0 = floorf(gx);
    float wy = gy - fy0, wx = gx - fx0;
    int y0 = (int)fy0, x0 = (int)fx0;
    int y1 = min(y0 + 1, HH - 1), x1 = min(x0 + 1, WW - 1);
    float w00 = (1.f - wy) * (1.f - wx), w01 = (1.f - wy) * wx;
    float w10 = wy * (1.f - wx),         w11 = wy * wx;

    float* base = gdc + ((size_t)(ph * ZE + ze) * HWSZ) * 2;
    atomicAdd(base + (size_t)(y0 * WW + x0) * 2,     w00 * gr);
    atomicAdd(base + (size_t)(y0 * WW + x0) * 2 + 1, w00 * gi);
    atomicAdd(base + (size_t)(y0 * WW + x1) * 2,     w01 * gr);
    atomicAdd(base + (size_t)(y0 * WW + x1) * 2 + 1, w01 * gi);
    atomicAdd(base + (size_t)(y1 * WW + x0) * 2,     w10 * gr);
    atomicAdd(base + (size_t)(y1 * WW + x0) * 2 + 1, w10 * gi);
    atomicAdd(base + (size_t)(y1 * WW + x1) * 2,     w11 * gr);
    atomicAdd(base + (size_t)(y1 * WW + x1) * 2 + 1, w11 * gi);
}

// ---------------------------------------------------------------------------
// TV gradient + gradient-descent update (writes xnew to avoid neighbor races)
// ---------------------------------------------------------------------------
__global__ void k_tv_update(const float* __restrict__ xcur,
                            const float* __restrict__ gdc,
                            const float* __restrict__ stdv,
                            float* __restrict__ xnew) {
    int idx = blockIdx.x * blockDim.x + threadIdx.x;
    if (idx >= PHN * ZE * HWSZ) return;
    int ph = idx / (ZE * HWSZ);
    int r  = idx % (ZE * HWSZ);
    int ze = r / HWSZ;
    int p  = r % HWSZ;
    int z  = ze + 1;
    int y  = p / WW, x = p % WW;

    const float* base = xcur + ((size_t)(ph * ZF + z) * HWSZ) * 2;
    float2 c = *(const float2*)(base + (size_t)p * 2);

    float tr = 0.0f, ti = 0.0f;
    if (y > 0) {            // head of up-edge: e = c - up
        float2 u = *(const float2*)(base + (size_t)((y - 1) * WW + x) * 2);
        float er = c.x - u.x, ei = c.y - u.y;
        float rr = rsqrtf(er * er + ei * ei + 1e-8f);
        tr += er * rr; ti += ei * rr;
    }
    if (y < HH - 1) {       // tail of down-edge: e = down - c
        float2 d = *(const float2*)(base + (size_t)((y + 1) * WW + x) * 2);
        float er = d.x - c.x, ei = d.y - c.y;
        float rr = rsqrtf(er * er + ei * ei + 1e-8f);
        tr -= er * rr; ti -= ei * rr;
    }
    if (x > 0) {
        float2 l = *(const float2*)(base + (size_t)(y * WW + x - 1) * 2);
        float er = c.x - l.x, ei = c.y - l.y;
        float rr = rsqrtf(er * er + ei * ei + 1e-8f);
        tr += er * rr; ti += ei * rr;
    }
    if (x < WW - 1) {
        float2 rv = *(const float2*)(base + (size_t)(y * WW + x + 1) * 2);
        float er = rv.x - c.x, ei = rv.y - c.y;
        float rr = rsqrtf(er * er + ei * ei + 1e-8f);
        tr -= er * rr; ti -= ei * rr;
    }

    float s = stdv[0];
    float2 g = *(const float2*)(gdc + ((size_t)(ph * ZE + ze) * HWSZ + p) * 2);
    float nr = c.x - GAMMA * g.x - TAU * s * tr;
    float ni = c.y - GAMMA * g.y - TAU * s * ti;
    *(float2*)(xnew + ((size_t)(ph * ZE + ze) * HWSZ + p) * 2) = make_float2(nr, ni);
}

__global__ void k_copyback(const float* __restrict__ xnew, float* __restrict__ xcur) {
    int idx = blockIdx.x * blockDim.x + threadIdx.x;
    if (idx >= PHN * ZE * HWSZ) return;
    int ph = idx / (ZE * HWSZ);
    int r  = idx % (ZE * HWSZ);
    int ze = r / HWSZ;
    int p  = r % HWSZ;
    int z  = ze + 1;
    float2 v = *(const float2*)(xnew + ((size_t)(ph * ZE + ze) * HWSZ + p) * 2);
    *(float2*)(xcur + ((size_t)(ph * ZF + z) * HWSZ + p) * 2) = v;
}

__global__ void k_final(const float* __restrict__ xcur, float* __restrict__ out,
                        long long n2) {
    long long i = (long long)blockIdx.x * blockDim.x + threadIdx.x;
    if (i < n2) out[i] = xcur[i];
}

// ---------------------------------------------------------------------------
// Host orchestration (graph-capture safe: only kernel launches on `stream`)
// ---------------------------------------------------------------------------
extern "C" void kernel_launch(void* const* d_in, const int* in_sizes, int n_in,
                              void* d_out, int out_size, void* d_ws, size_t ws_size,
                              hipStream_t stream) {
    (void)in_sizes; (void)n_in; (void)out_size; (void)ws_size;
    const float* kdata = (const float*)d_in[0];   // (1,PH,CH,Z,NK,2)
    const float* traj  = (const float*)d_in[1];   // (PH,2,NK)
    const float* img0  = (const float*)d_in[2];   // (1,PH,Z,H,W,2)
    const float* mvf   = (const float*)d_in[3];   // (PH,2,Z,H,W)
    const float* csm   = (const float*)d_in[4];   // (CH,Z,H,W,2)
    const float* stdv  = (const float*)d_in[5];   // (1,)
    float* out = (float*)d_out;

    // Workspace layout (floats).  Total ~98.6M floats (~394 MB).
    float* ws = (float*)d_ws;
    const long long SLICEF = (long long)HWSZ * 2;            // floats per slice
    float* D     = ws;                                       // 204800
    float* Dc    = D    + (long long)HWSZ * 2;               // 204800
    float* bufA  = Dc   + (long long)HWSZ * 2;               // 192 slices
    float* bufB  = bufA + (long long)NSLICE * SLICEF;
    float* est   = bufB + (long long)NSLICE * SLICEF;        // (192,NK) c
    float* res   = est  + (long long)NSLICE * NK * 2;
    float* gdc   = res  + (long long)NSLICE * NK * 2;        // (PH,ZE,H,W) c
    float* xnew  = gdc  + (long long)PHN * ZE * HWSZ * 2;
    float* xcur  = xnew + (long long)PHN * ZE * HWSZ * 2;    // (PH,ZF,H,W) c
    float* maxsq = xcur + (long long)PHN * ZF * HWSZ * 2;    // 1

    auto cdiv = [](long long a, long long b) { return (int)((a + b - 1) / b); };
    const long long nEff  = (long long)PHN * ZE * HWSZ;
    const long long nFull = (long long)PHN * ZF * HWSZ * 2;
    const long long nKsp  = (long long)NSLICE * NK;
    const dim3 gemmGrid(2, 40, NSLICE);          // 160-row stripes x 8-ccol groups

    k_twiddle<<<cdiv(HWSZ, 256), 256, 0, stream>>>(D, Dc);
    k_init_x<<<cdiv(nFull, 256), 256, 0, stream>>>(img0, xcur, nFull);

    for (int it = 0; it < NITER; ++it) {
        // ---- forward model ----
        k_warp_coil<<<cdiv(nEff, 256), 256, 0, stream>>>(xcur, mvf, csm, bufA);
        // T^T = (X*D)^T
        k_cgemm320<<<gemmGrid, 320, 0, stream>>>(bufA, SLICEF, D, bufB, SLICEF, 1);
        // F = (T^T*D)^T  (D symmetric => F = D*T = fft2)
        k_cgemm320<<<gemmGrid, 320, 0, stream>>>(bufB, SLICEF, D, bufA, SLICEF, 1);
        k_sample<<<cdiv(nKsp, 256), 256, 0, stream>>>(bufA, traj, est);

        // ---- weights + residual ----
        k_fill<<<1, 64, 0, stream>>>(maxsq, 1);
        k_maxabs<<<512, 256, 0, stream>>>(est, nKsp, maxsq);
        k_residual<<<cdiv(nKsp, 256), 256, 0, stream>>>(est, kdata, maxsq, res);

        // ---- adjoint chain ----
        k_fill<<<4096, 256, 0, stream>>>(bufB, (long long)NSLICE * SLICEF);
        k_scatter<<<cdiv(nKsp, 256), 256, 0, stream>>>(res, traj, bufB);   // G^T
        // P = (G^T*Dc)^T = Dc*G
        k_cgemm320<<<gemmGrid, 320, 0, stream>>>(bufB, SLICEF, Dc, bufA, SLICEF, 1);
        // Gimg = P*Dc = Dc*G*Dc   (normal store -> coalesced combine reads)
        k_cgemm320<<<gemmGrid, 320, 0, stream>>>(bufA, SLICEF, Dc, bufB, SLICEF, 0);
        k_fill<<<2048, 256, 0, stream>>>(gdc, nEff * 2);
        k_combine_warpadj<<<cdiv(nEff, 256), 256, 0, stream>>>(bufB, csm, mvf, gdc);

        // ---- TV + update ----
        k_tv_update<<<cdiv(nEff, 256), 256, 0, stream>>>(xcur, gdc, stdv, xnew);
        k_copyback<<<cdiv(nEff, 256), 256, 0, stream>>>(xnew, xcur);
    }

    k_final<<<cdiv(nFull, 256), 256, 0, stream>>>(xcur, out, nFull);
}